// DeployModel_12249246729013
// MI455X (gfx1250) — compile-verified
//
#include <hip/hip_runtime.h>
#include <hip/hip_bf16.h>
#include <stdint.h>

// ---------------- problem constants (match reference) ----------------
#define NUM_CLASSES 80
#define BATCH       16
#define A_TOTAL     8400          // 6400 + 1600 + 400
#define PRE_TOP_K   1000
#define PAD_K       1024          // pre_top_k padded to pow2 for tiles/scan
#define SORT_N      16384         // 8400 padded to pow2 for bitonic sort
#define KEEP_TOP_K  100
#define IOU_THR     0.65f
#define SCORE_THR   0.25f
#define CLASS_OFF   8192.0f

typedef __attribute__((ext_vector_type(2))) float v2f;
typedef __attribute__((ext_vector_type(8))) float v8f;

#if defined(__has_builtin)
#  if __has_builtin(__builtin_amdgcn_wmma_f32_16x16x4_f32)
#    define WMMA_F32X4 1
#  endif
#endif
#ifndef WMMA_F32X4
#  define WMMA_F32X4 0
#endif

// =====================================================================
// Kernel 1: per-anchor class max (sigmoid(max) == max(sigmoid)) + decode
// =====================================================================
__global__ __launch_bounds__(256) void decode_kernel(
    const float* __restrict__ cls0, const float* __restrict__ cls1,
    const float* __restrict__ cls2, const float* __restrict__ box0,
    const float* __restrict__ box1, const float* __restrict__ box2,
    float* __restrict__ score_all, int* __restrict__ label_all,
    float* __restrict__ box_all) {
  int gid = blockIdx.x * blockDim.x + threadIdx.x;
  if (gid >= BATCH * A_TOTAL) return;
  int b = gid / A_TOTAL;
  int a = gid - b * A_TOTAL;

  const float* cls; const float* boxp; int hw, la, W; float s;
  if (a < 6400)      { cls = cls0; boxp = box0; hw = 6400; la = a;        s = 8.f;  W = 80; }
  else if (a < 8000) { cls = cls1; boxp = box1; hw = 1600; la = a - 6400; s = 16.f; W = 40; }
  else               { cls = cls2; boxp = box2; hw = 400;  la = a - 8000; s = 32.f; W = 20; }

  // class max + argmax (first occurrence) — coalesced: adjacent threads share la+1
  const float* cp = cls + (size_t)b * NUM_CLASSES * hw + la;
  float best = cp[0];
  int bi = 0;
  for (int c = 1; c < NUM_CLASSES; ++c) {
    float v = cp[(size_t)c * hw];
    if (v > best) { best = v; bi = c; }
  }
  float score = 1.0f / (1.0f + expf(-best));

  // rtmdet decode: priors (x = w*s, y = h*s), distances * stride
  float px = (float)(la % W) * s;
  float py = (float)(la / W) * s;
  const float* bp = boxp + (size_t)b * 4 * hw + la;
  float d0 = bp[0] * s, d1 = bp[(size_t)hw] * s;
  float d2 = bp[(size_t)2 * hw] * s, d3 = bp[(size_t)3 * hw] * s;

  score_all[gid] = score;
  label_all[gid] = bi;
  float* ob = box_all + (size_t)gid * 4;
  ob[0] = px - d0; ob[1] = py - d1; ob[2] = px + d2; ob[3] = py + d3;
}

// =====================================================================
// Kernel 2: per-image top-1000 via full bitonic sort of packed keys in LDS
// key = (float_bits(masked_score) << 32) | (0xFFFFFFFF - index)
//   -> descending sort == score desc, tie lower index first (top_k semantics)
// =====================================================================
__global__ __launch_bounds__(1024) void topk_kernel(
    const float* __restrict__ score_all, const int* __restrict__ label_all,
    const float* __restrict__ box_all, float* __restrict__ tk_score,
    int* __restrict__ tk_label, float* __restrict__ tk_box) {
  extern __shared__ unsigned long long keys[];  // SORT_N * 8B = 128 KB
  int b = blockIdx.x, tid = threadIdx.x;

  for (int i = tid; i < SORT_N; i += 1024) {
    unsigned long long k = 0ull;
    if (i < A_TOTAL) {
      float sc = score_all[(size_t)b * A_TOTAL + i];
      float m = (sc > SCORE_THR) ? sc : 0.0f;   // pre-topk masking (reference)
      k = ((unsigned long long)__float_as_uint(m) << 32) |
          (unsigned long long)(0xFFFFFFFFu - (unsigned)i);
    }
    keys[i] = k;
  }
  __syncthreads();

  for (int k = 2; k <= SORT_N; k <<= 1) {
    for (int j = k >> 1; j > 0; j >>= 1) {
      for (int i = tid; i < SORT_N; i += 1024) {
        int ixj = i ^ j;
        if (ixj > i) {
          unsigned long long x = keys[i], y = keys[ixj];
          bool up = ((i & k) == 0);                // descending blocks
          bool sw = up ? (x < y) : (x > y);
          if (sw) { keys[i] = y; keys[ixj] = x; }
        }
      }
      __syncthreads();
    }
  }

  for (int p = tid; p < PRE_TOP_K; p += 1024) {
    unsigned long long k = keys[p];
    float sc = __uint_as_float((unsigned)(k >> 32));
    unsigned a = 0xFFFFFFFFu - (unsigned)(k & 0xFFFFFFFFull);
    size_t src = (size_t)b * A_TOTAL + a;
    size_t dst = (size_t)b * PAD_K + p;
    tk_score[dst] = sc;
    tk_label[dst] = label_all[src];
    const float* sb = box_all + src * 4;
    float* db = tk_box + dst * 4;
    db[0] = sb[0]; db[1] = sb[1]; db[2] = sb[2]; db[3] = sb[3];
  }
}

// =====================================================================
// Kernel 3: per-image NMS.
//  - LDS: 1024x1024 suppression bit matrix (128 KB) + offset boxes + areas.
//  - IoU tiles 16x16: pairwise (area_i + area_j) via V_WMMA_F32_16X16X4_F32
//    rank-2 outer sum; intersection via VALU; iou>thr -> ballot -> ds_or.
//  - wave0 sequential greedy scan (32 lanes own one 32-bit word each).
// =====================================================================
__global__ __launch_bounds__(1024) void nms_kernel(
    const float* __restrict__ tk_score, const int* __restrict__ tk_label,
    const float* __restrict__ tk_box, float* __restrict__ out) {
  extern __shared__ char smem[];
  unsigned* bits  = (unsigned*)smem;                              // PAD_K*32 words
  float4*   obox  = (float4*)(smem + (size_t)PAD_K * 32 * 4);     // 16 KB
  float*    areas = (float*)((char*)obox + (size_t)PAD_K * 16);   // 4 KB
  float*    tsc   = areas + PAD_K;                                // 4 KB
  unsigned* keeps = (unsigned*)(tsc + PAD_K);                     // 32
  unsigned* excl  = keeps + 32;                                   // 32
  unsigned* totp  = excl + 32;                                    // 1

  int b = blockIdx.x, tid = threadIdx.x;

  for (int i = tid; i < PAD_K * 32; i += 1024) bits[i] = 0u;
  if (tid < PAD_K) {
    if (tid < PRE_TOP_K) {
      size_t src = (size_t)b * PAD_K + tid;
      float sc = tk_score[src];
      float off = (float)tk_label[src] * CLASS_OFF;   // batched_nms class offset
      const float* bp = tk_box + src * 4;
      float4 ob = make_float4(bp[0] + off, bp[1] + off, bp[2] + off, bp[3] + off);
      obox[tid] = ob;
      areas[tid] = fmaxf(ob.z - ob.x, 0.f) * fmaxf(ob.w - ob.y, 0.f);
      tsc[tid] = sc;
    } else {                                          // inert padding rows/cols
      float far = -1.0e9f - (float)tid * 1024.0f;
      obox[tid] = make_float4(far, far, far, far);
      areas[tid] = 0.f;
      tsc[tid] = 0.f;
    }
  }
  // zero-init this image's output region (fill slots may be sparse)
  if (tid < KEEP_TOP_K * 5) out[(size_t)b * KEEP_TOP_K * 5 + tid] = 0.f;
  if (tid < KEEP_TOP_K)
    out[(size_t)BATCH * KEEP_TOP_K * 5 + (size_t)b * KEEP_TOP_K + tid] = 0.f;
  __threadfence_block();
  __syncthreads();

  // ---- IoU bit-matrix: 64x64 tiles of 16x16, 32 waves round-robin ----
  int wave = tid >> 5, lane = tid & 31;
  int hi = lane >> 4;                 // 0: lanes 0-15, 1: lanes 16-31
  for (int t = wave; t < 64 * 64; t += 32) {
    int ti = t >> 6, tj = t & 63;
    int i0 = ti * 16, j0 = tj * 16;
    int ncol = j0 + (lane & 15);      // this lane's column (both half-waves)
    float4 cb = obox[ncol];
    float carea = areas[ncol];

    float sums[8];
#if WMMA_F32X4
    // A(16x4): col0 = area_row, col1 = 1 ; B(4x16): row0 = 1, row1 = area_col
    // => C[m][n] = area_row[m] + area_col[n]
    v2f av, bv;
    if (hi == 0) { av.x = areas[i0 + lane]; av.y = 1.0f; bv.x = 1.0f; bv.y = carea; }
    else         { av.x = 0.0f; av.y = 0.0f; bv.x = 0.0f; bv.y = 0.0f; }  // K=2,3
    v8f cf = {};
    cf = __builtin_amdgcn_wmma_f32_16x16x4_f32(false, av, false, bv,
                                               (short)0, cf, false, false);
#pragma unroll
    for (int v = 0; v < 8; ++v) sums[v] = cf[v];
#else
#pragma unroll
    for (int v = 0; v < 8; ++v) sums[v] = areas[i0 + v + (hi << 3)] + carea;
#endif

#pragma unroll
    for (int v = 0; v < 8; ++v) {
      int m = i0 + v + (hi << 3);     // C layout: VGPR v -> M=v (lo) / v+8 (hi)
      float4 rb = obox[m];
      float iw = fmaxf(fminf(rb.z, cb.z) - fmaxf(rb.x, cb.x), 0.f);
      float ih = fmaxf(fminf(rb.w, cb.w) - fmaxf(rb.y, cb.y), 0.f);
      float inter = iw * ih;
      float uni = fmaxf(sums[v] - inter, 1e-6f);
      bool pred = inter > IOU_THR * uni;              // iou > thr
      unsigned bl = (unsigned)__ballot(pred);         // wave32: 32 valid bits
      if (lane == 0) {
        int word = j0 >> 5, sh = j0 & 16;
        atomicOr(&bits[(i0 + v) * 32 + word], (bl & 0xFFFFu) << sh);
        atomicOr(&bits[(i0 + v + 8) * 32 + word], (bl >> 16) << sh);
      }
    }
  }
  __syncthreads();

  // ---- sequential greedy scan, wave 0; lane w owns suppressed word w ----
  if (tid < 32) {
    unsigned sup = 0u;                                // init: ~valid
    for (int t2 = 0; t2 < 32; ++t2) {
      int j = tid * 32 + t2;
      if (!(tsc[j] > SCORE_THR)) sup |= (1u << t2);
    }
    unsigned keepw = 0u;
    for (int i = 0; i < PRE_TOP_K; ++i) {
      int w = i >> 5, bpos = i & 31;
      unsigned swi = (unsigned)__shfl((int)sup, w);   // broadcast word w
      bool keep = ((swi >> bpos) & 1u) == 0u;         // uniform across wave
      if (keep) {
        sup |= bits[i * 32 + tid];                    // suppress overlaps
        if (tid == w) {
          sup &= ~(1u << bpos);                       // keep self (diag)
          keepw |= (1u << bpos);
        }
      }
    }
    keeps[tid] = keepw;
  }
  __syncthreads();
  if (tid == 0) {
    unsigned run = 0;
    for (int w = 0; w < 32; ++w) { excl[w] = run; run += (unsigned)__popc(keeps[w]); }
    *totp = run;
  }
  __syncthreads();

  // ---- final top-100: kept (already score-sorted) then non-kept asc ----
  if (tid < PRE_TOP_K) {
    int w = tid >> 5, bpos = tid & 31;
    unsigned kw = keeps[w];
    bool kept = (kw >> bpos) & 1u;
    int rank = (int)excl[w] + __popc(kw & ((1u << bpos) - 1u));
    int total = (int)*totp;
    int tk = total < KEEP_TOP_K ? total : KEEP_TOP_K;
    int slot = -1;
    float sc = 0.f;
    if (kept) {
      if (rank < KEEP_TOP_K) { slot = rank; sc = tsc[tid]; }
    } else {
      int nk = tid - rank;                            // non-kept before tid
      slot = tk + nk;
      if (slot >= KEEP_TOP_K) slot = -1;
    }
    if (slot >= 0) {
      size_t src = (size_t)b * PAD_K + tid;
      const float* bp = tk_box + src * 4;             // un-offset boxes
      float* dp = out + (size_t)b * KEEP_TOP_K * 5 + (size_t)slot * 5;
      dp[0] = bp[0]; dp[1] = bp[1]; dp[2] = bp[2]; dp[3] = bp[3]; dp[4] = sc;
      out[(size_t)BATCH * KEEP_TOP_K * 5 + (size_t)b * KEEP_TOP_K + slot] =
          (float)tk_label[src];
    }
  }
}

// =====================================================================
extern "C" void kernel_launch(void* const* d_in, const int* in_sizes, int n_in,
                              void* d_out, int out_size, void* d_ws,
                              size_t ws_size, hipStream_t stream) {
  (void)in_sizes; (void)n_in; (void)out_size; (void)ws_size;
  const float* cls0 = (const float*)d_in[0];
  const float* cls1 = (const float*)d_in[1];
  const float* cls2 = (const float*)d_in[2];
  const float* box0 = (const float*)d_in[3];
  const float* box1 = (const float*)d_in[4];
  const float* box2 = (const float*)d_in[5];
  float* out = (float*)d_out;

  // workspace layout (floats): ~3.62 MB total
  float* ws        = (float*)d_ws;
  float* score_all = ws;                                        // B*A
  int*   label_all = (int*)(ws + (size_t)BATCH * A_TOTAL);      // B*A
  float* box_all   = ws + (size_t)2 * BATCH * A_TOTAL;          // B*A*4
  float* tk_score  = box_all + (size_t)BATCH * A_TOTAL * 4;     // B*PAD_K
  int*   tk_label  = (int*)(tk_score + (size_t)BATCH * PAD_K);  // B*PAD_K
  float* tk_box    = tk_score + (size_t)2 * BATCH * PAD_K;      // B*PAD_K*4

  decode_kernel<<<(BATCH * A_TOTAL + 255) / 256, 256, 0, stream>>>(
      cls0, cls1, cls2, box0, box1, box2, score_all, label_all, box_all);

  size_t smem_sort = (size_t)SORT_N * sizeof(unsigned long long);   // 128 KB
  topk_kernel<<<BATCH, 1024, smem_sort, stream>>>(
      score_all, label_all, box_all, tk_score, tk_label, tk_box);

  size_t smem_nms = (size_t)PAD_K * 32 * 4     // bit matrix 128 KB
                  + (size_t)PAD_K * 16         // offset boxes
                  + (size_t)PAD_K * 4 * 2      // areas + scores
                  + 65 * 4 + 64;               // keeps/excl/total + slack
  nms_kernel<<<BATCH, 1024, smem_nms, stream>>>(
      tk_score, tk_label, tk_box, out);
}